// Spikeslab_GPsampler_19696720019887
// MI455X (gfx1250) — compile-verified
//
#include <hip/hip_runtime.h>
#include <math.h>

#define NB   8
#define ND   768
#define NCH  3
#define NMIX 5
#define NS   10
#define NH   10
#define NTILE 3                    // ND / 256 i-tiles per (b,c)
#define PI2f 6.28318530717958647692f
#define ZITTERf 1e-4f
#define TEMPf 0.1f
#define GEPSf 1e-20f

typedef float v2f __attribute__((ext_vector_type(2)));
typedef float v8f __attribute__((ext_vector_type(8)));

// ---------------- workspace layout (floats) ----------------
// [0, 720)    : hpart[(b*NCH+c)*NTILE + itile][NH]  partial sums of relu(W1@tif+b1) over i
// [768, 888)  : wout[b][c][m]   mixture weights (logits_s)
// [896, 899)  : diag[c] = ZITTER + clip(likerr)^2
#define WS_HPART 0
#define WS_WOUT  768
#define WS_DIAG  896

// =====================================================================
// Kernel 1: fused SM-kernel generation + GEMV + first MLP layer + block
// reduction over i.  grid = NB*NCH*NTILE blocks of 256 threads.
// =====================================================================
__global__ void k_feature(const float* __restrict__ xc, const float* __restrict__ yc,
                          const float* __restrict__ mu, const float* __restrict__ inv_std,
                          const float* __restrict__ W1, const float* __restrict__ b1,
                          float* __restrict__ ws)
{
    __shared__ float sxj[ND];
    __shared__ float syj[ND];
    __shared__ float red[256][NH + 1];          // +1 pad vs bank conflicts

    const int tid = threadIdx.x;
    int blk = blockIdx.x;
    const int itile = blk % NTILE;  blk /= NTILE;
    const int c     = blk % NCH;    blk /= NCH;
    const int b     = blk;
    const int i     = itile * 256 + tid;

    // stage x, y for all j of this (b, c)
    for (int j = tid; j < ND; j += 256) {
        sxj[j] = xc[(b * ND + j) * NCH + c];    // xc[b, j, 0, c]
        syj[j] = yc[(b * ND + j) * NCH + c];
    }
    __syncthreads();

    float ke[NMIX], kc[NMIX];
#pragma unroll
    for (int m = 0; m < NMIX; ++m) {
        const float is = inv_std[m];
        ke[m] = 0.5f * PI2f * PI2f * is * is;
        kc[m] = PI2f * mu[m];
    }

    const float xi = sxj[i];
    const float yi = syj[i];
    float acc[NMIX] = {0.f, 0.f, 0.f, 0.f, 0.f};

    for (int j = 0; j < ND; ++j) {
        const float d  = xi - sxj[j];
        const float d2 = d * d;
        const float yj = syj[j];
#pragma unroll
        for (int m = 0; m < NMIX; ++m)
            acc[m] = fmaf(__expf(-ke[m] * d2) * __cosf(kc[m] * d), yj, acc[m]);
    }
#pragma unroll
    for (int m = 0; m < NMIX; ++m) acc[m] += ZITTERf * yi;   // + zitter * yc

    // hh1 = relu(tif @ W1^T + b1), tif = [feature(5), yc]
#pragma unroll
    for (int k = 0; k < NH; ++k) {
        float v = b1[k];
#pragma unroll
        for (int m = 0; m < NMIX; ++m) v = fmaf(W1[k * (NMIX + 1) + m], acc[m], v);
        v = fmaf(W1[k * (NMIX + 1) + NMIX], yi, v);
        red[tid][k] = fmaxf(v, 0.0f);
    }
    __syncthreads();

    // deterministic tree reduction over the 256 i's of this tile
    for (int s = 128; s > 0; s >>= 1) {
        if (tid < s) {
#pragma unroll
            for (int k = 0; k < NH; ++k) red[tid][k] += red[tid + s][k];
        }
        __syncthreads();
    }
    if (tid < NH)
        ws[WS_HPART + blockIdx.x * NH + tid] = red[0][tid];
}

// =====================================================================
// Kernel 2: one wave32.  MLP via V_WMMA_F32_16X16X4_F32, then Gumbel-
// softmax mixture weights + diagonal constants.
// =====================================================================
__device__ __forceinline__ void wmma_layer(const float (*X)[32], float (*Y)[32],
                                           const float* __restrict__ W,
                                           const float* __restrict__ bias,
                                           int N, int K, bool do_relu, int lane)
{
    v8f c = {};                                 // f32 accumulator, 8 VGPRs
    const int lr    = lane & 15;                // M row (A) / N col (B)
    const int khalf = (lane >> 4) << 1;         // lanes 0-15 -> K 0,1 ; 16-31 -> K 2,3

    for (int k0 = 0; k0 < K; k0 += 4) {         // uniform loop, EXEC stays full
        const int ka = k0 + khalf;
        v2f a, bm;
        a.x  = X[lr][ka];                       // X is zero-padded to 16x32
        a.y  = X[lr][ka + 1];
        bm.x = (lr < N && ka     < K) ? W[lr * K + ka]     : 0.0f;  // B[k][n] = W[n][k]
        bm.y = (lr < N && ka + 1 < K) ? W[lr * K + ka + 1] : 0.0f;
        c = __builtin_amdgcn_wmma_f32_16x16x4_f32(false, a, false, bm,
                                                  (short)0, c, false, false);
    }

    // zero destination buffer (keeps padding regions zero for next layer)
    for (int t = lane; t < 16 * 32; t += 32) ((float*)Y)[t] = 0.0f;
    __syncthreads();

    // lanes 0-15 hold rows 0..7 of column `lane` in c[0..7]; M=8 so that's all we need
    if (lane < 16 && lane < N) {
#pragma unroll
        for (int r = 0; r < 8; ++r) {
            float v = c[r] + bias[lane];
            if (do_relu) v = fmaxf(v, 0.0f);
            Y[r][lane] = v;
        }
    }
    __syncthreads();
}

__global__ void k_mlp(const float* __restrict__ W2, const float* __restrict__ b2,
                      const float* __restrict__ W3, const float* __restrict__ b3,
                      const float* __restrict__ W4, const float* __restrict__ b4,
                      const float* __restrict__ W5, const float* __restrict__ b5,
                      const float* __restrict__ likerr, const float* __restrict__ unif,
                      float* __restrict__ ws)
{
    __shared__ float A[16][32];
    __shared__ float B[16][32];
    const int lane = threadIdx.x;

    // X0[b][c*NH+k] = mean over i of hh1 = (sum of NTILE partials) / ND
    for (int t = lane; t < 16 * 32; t += 32) ((float*)A)[t] = 0.0f;
    __syncthreads();
    for (int t = lane; t < NB * NCH * NH; t += 32) {
        const int bb  = t / (NCH * NH);
        const int rem = t % (NCH * NH);          // = c*NH + k  (column index)
        const int bc  = bb * NCH + rem / NH;
        const int k   = rem % NH;
        float s = 0.f;
#pragma unroll
        for (int tt = 0; tt < NTILE; ++tt)
            s += ws[WS_HPART + (bc * NTILE + tt) * NH + k];
        A[bb][rem] = s * (1.0f / (float)ND);
    }
    __syncthreads();

    wmma_layer(A, B, W2, b2, NH,         NCH * NH, true,  lane);  // (8x30)@(30x10)
    wmma_layer(B, A, W3, b3, NH,         NH,       true,  lane);  // (8x10)@(10x10)
    wmma_layer(A, B, W4, b4, NH,         NH,       true,  lane);  // (8x10)@(10x10)
    wmma_layer(B, A, W5, b5, NCH * NMIX, NH,       false, lane);  // (8x10)@(10x15) -> A = loglogits

    // Gumbel-softmax: one (b,c) per lane, sequential over s => deterministic
    if (lane < NB * NCH) {
        const int bb = lane / NCH;
        const int cc = lane % NCH;
        float accm[NMIX] = {0.f, 0.f, 0.f, 0.f, 0.f};
        for (int s = 0; s < NS; ++s) {
            float v[NMIX];
            float mx = -1e30f;
#pragma unroll
            for (int m = 0; m < NMIX; ++m) {
                const float u = unif[((bb * NS + s) * NCH + cc) * NMIX + m];
                const float g = -__logf(-__logf(u + GEPSf));
                v[m] = (g + A[bb][cc * NMIX + m]) * (1.0f / TEMPf);
                mx = fmaxf(mx, v[m]);
            }
            float sum = 0.f;
#pragma unroll
            for (int m = 0; m < NMIX; ++m) { v[m] = __expf(v[m] - mx); sum += v[m]; }
            const float inv = 1.0f / sum;
#pragma unroll
            for (int m = 0; m < NMIX; ++m) accm[m] += v[m] * inv;
        }
#pragma unroll
        for (int m = 0; m < NMIX; ++m)
            ws[WS_WOUT + (bb * NCH + cc) * NMIX + m] = accm[m] * (1.0f / (float)NS);
    }
    if (lane < NCH) {
        float l = likerr[lane];
        l = fminf(fmaxf(l, 0.1f), 1.0f);
        ws[WS_DIAG + lane] = ZITTERf + l * l;
    }
}

// =====================================================================
// Kernel 3: weighted mixture reduction, streamed to the 56.6 MB output.
// grid = NB * (ND*ND/256) blocks of 256; one (b,i,j) per thread.
// =====================================================================
__global__ void k_out(const float* __restrict__ xc,
                      const float* __restrict__ mu, const float* __restrict__ inv_std,
                      const float* __restrict__ ws, float* __restrict__ out)
{
    const int tid = threadIdx.x;
    const int blocks_per_b = (ND * ND) / 256;
    const int b   = blockIdx.x / blocks_per_b;
    const int lin = (blockIdx.x % blocks_per_b) * 256 + tid;
    const int i   = lin / ND;
    const int j   = lin % ND;

    __builtin_prefetch(ws + WS_WOUT + b * NCH * NMIX, 0, 0);   // global_prefetch_b8

    float ke[NMIX], kc[NMIX];
#pragma unroll
    for (int m = 0; m < NMIX; ++m) {
        const float is = inv_std[m];
        ke[m] = 0.5f * PI2f * PI2f * is * is;
        kc[m] = PI2f * mu[m];
    }

    float res[NCH];
#pragma unroll
    for (int c = 0; c < NCH; ++c) {
        const float xi = xc[(b * ND + i) * NCH + c];
        const float xj = xc[(b * ND + j) * NCH + c];
        const float d  = xi - xj;
        const float d2 = d * d;
        float acc = 0.f;
#pragma unroll
        for (int m = 0; m < NMIX; ++m)
            acc = fmaf(ws[WS_WOUT + (b * NCH + c) * NMIX + m],
                       __expf(-ke[m] * d2) * __cosf(kc[m] * d), acc);
        if (i == j) acc += ws[WS_DIAG + c];
        res[c] = acc;
    }

    float* o = out + ((size_t)(b * ND + i) * ND + j) * NCH;
    o[0] = res[0]; o[1] = res[1]; o[2] = res[2];
}

// =====================================================================
extern "C" void kernel_launch(void* const* d_in, const int* in_sizes, int n_in,
                              void* d_out, int out_size, void* d_ws, size_t ws_size,
                              hipStream_t stream)
{
    const float* xc      = (const float*)d_in[0];
    const float* yc      = (const float*)d_in[1];
    const float* mu      = (const float*)d_in[2];
    const float* inv_std = (const float*)d_in[3];
    const float* likerr  = (const float*)d_in[4];
    const float* unif    = (const float*)d_in[5];
    const float* W1 = (const float*)d_in[6];  const float* b1 = (const float*)d_in[7];
    const float* W2 = (const float*)d_in[8];  const float* b2 = (const float*)d_in[9];
    const float* W3 = (const float*)d_in[10]; const float* b3 = (const float*)d_in[11];
    const float* W4 = (const float*)d_in[12]; const float* b4 = (const float*)d_in[13];
    const float* W5 = (const float*)d_in[14]; const float* b5 = (const float*)d_in[15];
    float* ws  = (float*)d_ws;
    float* out = (float*)d_out;

    k_feature<<<NB * NCH * NTILE, 256, 0, stream>>>(xc, yc, mu, inv_std, W1, b1, ws);
    k_mlp<<<1, 32, 0, stream>>>(W2, b2, W3, b3, W4, b4, W5, b5, likerr, unif, ws);
    k_out<<<NB * ((ND * ND) / 256), 256, 0, stream>>>(xc, mu, inv_std, ws, out);
}